// faBlockNew_53326313947742
// MI455X (gfx1250) — compile-verified
//
#include <hip/hip_runtime.h>
#include <hip/hip_bf16.h>

// ---------------- problem constants ----------------
#define BB 2
#define CC 64
#define DD 24
#define HH 96
#define WW 96
#define SS (DD*HH*WW)          // 221184 spatial per (b, c)
#define BSTRIDE (CC*SS)        // 14155776 elems per batch
#define NTOT (BB*(size_t)BSTRIDE)
#define KCH 1024               // K-chunk for energy partial sums

typedef __attribute__((ext_vector_type(2))) float v2f;
typedef __attribute__((ext_vector_type(8))) float v8f;

static __device__ __forceinline__ v8f wmma4(v2f a, v2f b, v8f c) {
  // D(16x16,f32) = A(16x4,f32) x B(4x16,f32) + C
  return __builtin_amdgcn_wmma_f32_16x16x4_f32(
      /*neg_a=*/false, a, /*neg_b=*/false, b,
      /*c_mod=*/(short)0, c, /*reuse_a=*/false, /*reuse_b=*/false);
}

// ---------------------------------------------------------------------------
// Kernel 1: fused three 1x1 convs: out[b,o,s] = W[o,c] x[b,c,s] + bias
// One wave per 16(o) x 16(s) tile, K = 64 in steps of 4; x-tile load shared
// across the three weight matrices (3 WMMAs per load).
// ---------------------------------------------------------------------------
__global__ __launch_bounds__(256) void conv3_kernel(
    const float* __restrict__ x,
    const float* __restrict__ wt, const float* __restrict__ bt,
    const float* __restrict__ wp, const float* __restrict__ bp,
    const float* __restrict__ wg, const float* __restrict__ bg,
    float* __restrict__ theta, float* __restrict__ phi, float* __restrict__ g,
    int total_tiles)
{
  int wid = blockIdx.x * (blockDim.x >> 5) + (threadIdx.x >> 5);
  if (wid >= total_tiles) return;
  const int s_tiles = SS / 16;
  int s_t = wid % s_tiles;
  int r   = wid / s_tiles;
  int o_t = r & 3;            // 64/16 = 4 output-channel tiles
  int b   = r >> 2;
  int lane = threadIdx.x & 31, lr = lane & 15, hi = lane >> 4;
  int o0 = o_t * 16, s0 = s_t * 16;
  const float* xb = x + (size_t)b * BSTRIDE;

  v8f ct = {0,0,0,0,0,0,0,0}, cp = ct, cg = ct;
  for (int k = 0; k < CC; k += 4) {
    int ka = k + hi * 2;
    v2f at, ap, ag, bx;
    at.x = wt[(o0 + lr) * CC + ka];     at.y = wt[(o0 + lr) * CC + ka + 1];
    ap.x = wp[(o0 + lr) * CC + ka];     ap.y = wp[(o0 + lr) * CC + ka + 1];
    ag.x = wg[(o0 + lr) * CC + ka];     ag.y = wg[(o0 + lr) * CC + ka + 1];
    bx.x = xb[(size_t)ka * SS + s0 + lr];
    bx.y = xb[(size_t)(ka + 1) * SS + s0 + lr];
    ct = wmma4(at, bx, ct);
    cp = wmma4(ap, bx, cp);
    cg = wmma4(ag, bx, cg);
  }
  for (int v = 0; v < 8; ++v) {
    int o = o0 + v + hi * 8;
    size_t idx = (size_t)b * BSTRIDE + (size_t)o * SS + s0 + lr;
    theta[idx] = ct[v] + bt[o];
    phi[idx]   = cp[v] + bp[o];
    g[idx]     = cg[v] + bg[o];
  }
}

// ---------------------------------------------------------------------------
// Kernel 2: attention energy: E[b,i,j] = sum_f t[b,i,f] * p[b,j,f]
// One wave = one 16-row i-tile x ALL TJ j-tiles (TJ accumulators): the A (t)
// load is shared across TJ WMMAs per K-step, cutting theta traffic TJ-fold.
// f enumerated by 3 nested dims; K split into chunks, partials via f32 atomics.
// ---------------------------------------------------------------------------
template<int TJ>
__global__ __launch_bounds__(256) void energy_kernel(
    const float* __restrict__ t, const float* __restrict__ p,
    float* __restrict__ e,
    int n, int sA, int n1, int n2, int st0, int st1, int st2,
    int F, int nchunks, int total_waves)
{
  int wid = blockIdx.x * (blockDim.x >> 5) + (threadIdx.x >> 5);
  if (wid >= total_waves) return;
  int chunk = wid % nchunks;
  int rest  = wid / nchunks;
  int it    = rest % TJ;       // i-tile (square: ti == TJ)
  int b     = rest / TJ;
  int lane = threadIdx.x & 31, lr = lane & 15, hi = lane >> 4;

  int kbeg = chunk * KCH;
  int kend = kbeg + KCH; if (kend > F) kend = F;
  // mixed-radix decode of kbeg -> (c0,c1,c2), then iterate with carries.
  int c12 = n1 * n2;
  int c0 = kbeg / c12; int r0 = kbeg - c0 * c12;
  int c1 = r0 / n2;    int c2 = r0 - c1 * n2;
  int off = c0 * st0 + c1 * st1 + c2 * st2;

  const float* tb = t + (size_t)b * BSTRIDE;
  const float* pb = p + (size_t)b * BSTRIDE;
  int i = it * 16 + lr;
  bool iok = i < n;
  int ibase = i * sA;
  int  jbase[TJ];
  bool jok[TJ];
#pragma unroll
  for (int jt = 0; jt < TJ; ++jt) {
    int j = jt * 16 + lr;
    jok[jt]   = j < n;
    jbase[jt] = j * sA;
  }

  v8f acc[TJ];
#pragma unroll
  for (int jt = 0; jt < TJ; ++jt) acc[jt] = (v8f){0,0,0,0,0,0,0,0};

  for (int k = kbeg; k < kend; k += 4) {
    int o0 = off + (hi ? 2 * st2 : 0);
    v2f a;
    a.x = iok ? tb[ibase + o0]       : 0.0f;
    a.y = iok ? tb[ibase + o0 + st2] : 0.0f;
#pragma unroll
    for (int jt = 0; jt < TJ; ++jt) {
      v2f bb;
      bb.x = jok[jt] ? pb[jbase[jt] + o0]       : 0.0f;
      bb.y = jok[jt] ? pb[jbase[jt] + o0 + st2] : 0.0f;
      acc[jt] = wmma4(a, bb, acc[jt]);
    }
    // advance f counters by 4 (4 divides n2 for every stage here)
    c2 += 4; off += 4 * st2;
    if (c2 == n2) {
      c2 = 0; off += st1 - n2 * st2; c1++;
      if (c1 == n1) { c1 = 0; off += st0 - n1 * st1; c0++; }
    }
  }
  float* eb = e + (size_t)b * n * n;
#pragma unroll
  for (int jt = 0; jt < TJ; ++jt) {
    int j = jt * 16 + lr;
#pragma unroll
    for (int v = 0; v < 8; ++v) {
      int ii = it * 16 + v + hi * 8;
      if (ii < n && j < n) atomicAdd(&eb[ii * n + j], acc[jt][v]);
    }
  }
}

// ---------------------------------------------------------------------------
// Kernel 3: per-row softmax of sign*E (sign=-1 implements softmax(rowmax-E)).
// rows = B*n (<=192), n <= 96 -> trivial one-thread-per-row.
// ---------------------------------------------------------------------------
__global__ void softmax_kernel(float* __restrict__ e, int n, int rows, float sign)
{
  int r = blockIdx.x * blockDim.x + threadIdx.x;
  if (r >= rows) return;
  float* row = e + (size_t)r * n;
  float m = -3.402823466e38f;
  for (int j = 0; j < n; ++j) { float v = sign * row[j]; m = fmaxf(m, v); }
  float s = 0.0f;
  for (int j = 0; j < n; ++j) { float v = expf(sign * row[j] - m); row[j] = v; s += v; }
  float inv = 1.0f / s;
  for (int j = 0; j < n; ++j) row[j] *= inv;
}

// ---------------------------------------------------------------------------
// Kernel 4: apply: dst[b,i,f] = src[b,i,f] + beta * sum_j A[b,i,j] src[b,j,f]
// One wave = one 16-wide f-block x ALL TI i-tiles (TI accumulators): the src
// column load (the only big-tensor read in the K loop) is shared across TI
// WMMAs, cutting src traffic TI-fold. attn (<=36KB) stays cache-hot.
// ---------------------------------------------------------------------------
template<int TI>
__global__ __launch_bounds__(256) void apply_kernel(
    const float* __restrict__ src, float* __restrict__ dst,
    const float* __restrict__ attn, const float* __restrict__ betas, int beta_idx,
    int n, int sA, int n1, int n2b, int st0, int st1, int st2,
    int Ft, int total_waves)
{
  int wid = blockIdx.x * (blockDim.x >> 5) + (threadIdx.x >> 5);
  if (wid >= total_waves) return;
  int ft = wid % Ft;
  int b  = wid / Ft;
  int lane = threadIdx.x & 31, lr = lane & 15, hi = lane >> 4;

  int blk2 = ft % n2b; int rr = ft / n2b;
  int c1 = rr % n1;    int c0 = rr / n1;
  int foff = c0 * st0 + c1 * st1 + (blk2 * 16 + lr) * st2;   // per-lane f offset

  const float* ab = attn + (size_t)b * n * n;
  const float* sb = src  + (size_t)b * BSTRIDE;
  float*       db = dst  + (size_t)b * BSTRIDE;

  v8f acc[TI];
#pragma unroll
  for (int it = 0; it < TI; ++it) acc[it] = (v8f){0,0,0,0,0,0,0,0};

  for (int k = 0; k < n; k += 4) {
    int ka = k + hi * 2;
    v2f bv;
    bv.x = sb[(size_t)ka * sA + foff];
    bv.y = sb[(size_t)(ka + 1) * sA + foff];
#pragma unroll
    for (int it = 0; it < TI; ++it) {
      int ia = it * 16 + lr;
      v2f a;
      a.x = (ia < n) ? ab[ia * n + ka]     : 0.0f;
      a.y = (ia < n) ? ab[ia * n + ka + 1] : 0.0f;
      acc[it] = wmma4(a, bv, acc[it]);
    }
  }
  float beta = betas[beta_idx];
#pragma unroll
  for (int it = 0; it < TI; ++it) {
#pragma unroll
    for (int v = 0; v < 8; ++v) {
      int ii = it * 16 + v + hi * 8;
      if (ii < n) {
        size_t idx = (size_t)ii * sA + foff;
        db[idx] = sb[idx] + beta * acc[it][v];
      }
    }
  }
}

// ---------------------------------------------------------------------------
extern "C" void kernel_launch(void* const* d_in, const int* in_sizes, int n_in,
                              void* d_out, int out_size, void* d_ws, size_t ws_size,
                              hipStream_t stream) {
  const float* x     = (const float*)d_in[0];
  const float* tw    = (const float*)d_in[1];
  const float* tb    = (const float*)d_in[2];
  const float* pw    = (const float*)d_in[3];
  const float* pb    = (const float*)d_in[4];
  const float* gw    = (const float*)d_in[5];
  const float* gb    = (const float*)d_in[6];
  const float* betas = (const float*)d_in[7];

  float* out    = (float*)d_out;
  float* theta  = (float*)d_ws;
  float* phi    = theta + NTOT;
  float* outB   = phi   + NTOT;
  float* energy = outB  + NTOT;

  // 1) fused convs: theta, phi in ws; g -> d_out (ping-pong start)
  {
    int tiles = BB * 4 * (SS / 16);          // 110592 waves
    conv3_kernel<<<tiles / 8, 256, 0, stream>>>(
        x, tw, tb, pw, pb, gw, gb, theta, phi, out, tiles);
  }

  struct Stage {
    const float* src; float* dst;
    int n, sA, n0, n1, n2, s0, s1, s2, F, bidx; float sign;
  };
  const Stage stages[4] = {
    // coronal  (H): f = (c,d,w)
    { out,  outB, HH, WW,      CC, DD, WW, SS,      HH*WW, 1,  CC*DD*WW, 0,  1.0f },
    // sagittal (W): f = (c,d,h)
    { outB, out,  WW, 1,       CC, DD, HH, SS,      HH*WW, WW, CC*DD*HH, 1,  1.0f },
    // axial    (D): f = (c,h,w)
    { out,  outB, DD, HH*WW,   CC, HH, WW, SS,      WW,    1,  CC*HH*WW, 2,  1.0f },
    // channel  (C): f = (d,h,w); softmax(rowmax-E) == softmax(-E)
    { outB, out,  CC, SS,      DD, HH, WW, HH*WW,   WW,    1,  DD*HH*WW, 3, -1.0f },
  };

  for (int s = 0; s < 4; ++s) {
    const Stage& st = stages[s];
    hipMemsetAsync(energy, 0, (size_t)BB * st.n * st.n * sizeof(float), stream);
    int ti = (st.n + 15) / 16;               // 6, 6, 2, 4
    int nchunks = (st.F + KCH - 1) / KCH;
    int ewaves = BB * ti * nchunks;
    int eblocks = (ewaves + 7) / 8;
    switch (ti) {
      case 2:
        energy_kernel<2><<<eblocks, 256, 0, stream>>>(
            theta, phi, energy, st.n, st.sA, st.n1, st.n2,
            st.s0, st.s1, st.s2, st.F, nchunks, ewaves);
        break;
      case 4:
        energy_kernel<4><<<eblocks, 256, 0, stream>>>(
            theta, phi, energy, st.n, st.sA, st.n1, st.n2,
            st.s0, st.s1, st.s2, st.F, nchunks, ewaves);
        break;
      default:
        energy_kernel<6><<<eblocks, 256, 0, stream>>>(
            theta, phi, energy, st.n, st.sA, st.n1, st.n2,
            st.s0, st.s1, st.s2, st.F, nchunks, ewaves);
        break;
    }
    softmax_kernel<<<1, 256, 0, stream>>>(energy, st.n, BB * st.n, st.sign);
    int n2b = st.n2 / 16;
    int Ft  = st.n0 * st.n1 * n2b;
    int awaves = BB * Ft;
    int ablocks = (awaves + 7) / 8;
    switch (ti) {
      case 2:
        apply_kernel<2><<<ablocks, 256, 0, stream>>>(
            st.src, st.dst, energy, betas, st.bidx, st.n, st.sA,
            st.n1, n2b, st.s0, st.s1, st.s2, Ft, awaves);
        break;
      case 4:
        apply_kernel<4><<<ablocks, 256, 0, stream>>>(
            st.src, st.dst, energy, betas, st.bidx, st.n, st.sA,
            st.n1, n2b, st.s0, st.s1, st.s2, Ft, awaves);
        break;
      default:
        apply_kernel<6><<<ablocks, 256, 0, stream>>>(
            st.src, st.dst, energy, betas, st.bidx, st.n, st.sA,
            st.n1, n2b, st.s0, st.s1, st.s2, Ft, awaves);
        break;
    }
  }
}